// EntityDetection_54013508715188
// MI455X (gfx1250) — compile-verified
//
#include <hip/hip_runtime.h>
#include <hip/hip_bf16.h>

#define B_ 512
#define T_ 512
#define L_ 64
#define BOS_ 62
#define EOS_ 63

typedef __attribute__((ext_vector_type(2))) float v2f;
typedef __attribute__((ext_vector_type(8))) float v8f;

union V8 { v8f v; float f[8]; float4 q[2]; };

// ---------------------------------------------------------------------------
// Forward algorithm: one wave handles 16 batches through the full T recurrence.
// Per step: S = exp(alpha - rowmax) @ exp(trans) via V_WMMA_F32_16X16X4_F32,
// then alpha' = em + rowmax + log(S), masked select.
// ---------------------------------------------------------------------------
__global__ __launch_bounds__(32)
void crf_forward_kernel(const float* __restrict__ em,
                        const float* __restrict__ trans,
                        const float* __restrict__ mask,
                        float* __restrict__ fwd_out) {
  __shared__ float Elds[L_ * L_];     // exp(trans), pair-interleaved B-layout
  __shared__ float aT[L_ * 16];       // transposed alphas: aT[j*16 + m]
  __shared__ float rmax_s[16];        // per-batch row max

  const int lane  = threadIdx.x;      // 0..31 (wave32)
  const int half  = lane >> 4;        // 0 or 1
  const int lq    = lane & 15;
  const int bbase = blockIdx.x * 16;

  // E[K][N] = exp(trans[K][N]) stored so the B-operand pair (K0,K0+1) for a
  // given lane is one contiguous b64: Elds[((K>>1)*64 + N)*2 + (K&1)]
  for (int idx = lane; idx < L_ * L_; idx += 32) {
    int K = idx >> 6, N = idx & 63;
    Elds[((K >> 1) * 64 + N) * 2 + (K & 1)] = __expf(trans[idx]);
  }

  float tB[4], tE[4];
#pragma unroll
  for (int n = 0; n < 4; ++n) {
    int j = 16 * n + lq;
    tB[n] = trans[BOS_ * L_ + j];     // transitions[BOS, j]
    tE[n] = trans[j * L_ + EOS_];     // transitions[j, EOS]
  }

  // alphas0 = trans[BOS,:] + emissions[:,0,:]  (C/D layout: M=v+8*half, N=lq)
  V8 al[4];
#pragma unroll
  for (int n = 0; n < 4; ++n) {
    int j = 16 * n + lq;
#pragma unroll
    for (int v = 0; v < 8; ++v) {
      int m = v + 8 * half;
      al[n].f[v] = tB[n] + em[((bbase + m) * T_ + 0) * L_ + j];
    }
  }
  __syncthreads();

  for (int t = 1; t < T_; ++t) {
    // Issue this step's global loads early (hide latency under LDS/VALU work).
    float emr[4][8];
#pragma unroll
    for (int n = 0; n < 4; ++n)
#pragma unroll
      for (int v = 0; v < 8; ++v)
        emr[n][v] = em[((bbase + v + 8 * half) * T_ + t) * L_ + 16 * n + lq];
    float mk[8];
#pragma unroll
    for (int v = 0; v < 8; ++v)
      mk[v] = mask[(bbase + v + 8 * half) * T_ + t];

    // Transpose alphas to LDS: aT[j*16 + m] (two ds_store_b128 per tile).
#pragma unroll
    for (int n = 0; n < 4; ++n) {
      int j = 16 * n + lq;
      float4* dst = (float4*)&aT[j * 16 + 8 * half];
      dst[0] = al[n].q[0];
      dst[1] = al[n].q[1];
    }
    __syncthreads();

    // A-layout read: lane owns batch m=lq; VGPR0->K=4k+2*half, VGPR1->K+1.
    float ar[16][2];
#pragma unroll
    for (int k = 0; k < 16; ++k) {
      int j0 = 4 * k + 2 * half;
      ar[k][0] = aT[(j0 + 0) * 16 + lq];
      ar[k][1] = aT[(j0 + 1) * 16 + lq];
    }
    float mx = -3.0e38f;
#pragma unroll
    for (int k = 0; k < 16; ++k) {
      mx = fmaxf(mx, ar[k][0]);
      mx = fmaxf(mx, ar[k][1]);
    }
    mx = fmaxf(mx, __shfl_xor(mx, 16, 32));   // combine lane-halves: full max
    rmax_s[lq] = mx;                          // both halves write same value
#pragma unroll
    for (int k = 0; k < 16; ++k) {
      ar[k][0] = __expf(ar[k][0] - mx);
      ar[k][1] = __expf(ar[k][1] - mx);
    }
    __syncthreads();

    float rmD[8];
#pragma unroll
    for (int v = 0; v < 8; ++v) rmD[v] = rmax_s[v + 8 * half];

    // GEMM: 4 N-tiles x 16 K-chunks of V_WMMA_F32_16X16X4_F32.
    V8 acc[4];
#pragma unroll
    for (int n = 0; n < 4; ++n)
#pragma unroll
      for (int v = 0; v < 8; ++v) acc[n].f[v] = 0.f;
#pragma unroll
    for (int k = 0; k < 16; ++k) {
      v2f A;
      A.x = ar[k][0];
      A.y = ar[k][1];
#pragma unroll
      for (int n = 0; n < 4; ++n) {
        v2f Bv = *(const v2f*)&Elds[((2 * k + half) * 64 + 16 * n + lq) * 2];
        acc[n].v = __builtin_amdgcn_wmma_f32_16x16x4_f32(
            false, A, false, Bv, (short)0, acc[n].v, false, false);
      }
    }

    // alpha' = em + rowmax + log(S); keep old alpha where mask == 0.
#pragma unroll
    for (int n = 0; n < 4; ++n)
#pragma unroll
      for (int v = 0; v < 8; ++v) {
        float na = emr[n][v] + rmD[v] + __logf(acc[n].f[v]);
        al[n].f[v] = (mk[v] > 0.f) ? na : al[n].f[v];
      }
    __syncthreads();
  }

  // end_scores = alphas + trans[:,EOS]; fwd[b] = logsumexp_j(end_scores)
#pragma unroll
  for (int n = 0; n < 4; ++n) {
    int j = 16 * n + lq;
    V8 fl;
#pragma unroll
    for (int v = 0; v < 8; ++v) fl.f[v] = al[n].f[v] + tE[n];
    float4* dst = (float4*)&aT[j * 16 + 8 * half];
    dst[0] = fl.q[0];
    dst[1] = fl.q[1];
  }
  __syncthreads();
  float mx = -3.0e38f;
  for (int j = 0; j < L_; ++j) mx = fmaxf(mx, aT[j * 16 + lq]);
  float s = 0.f;
  for (int j = 0; j < L_; ++j) s += __expf(aT[j * 16 + lq] - mx);
  fwd_out[bbase + lq] = mx + __logf(s);      // lane halves write identical data
}

// ---------------------------------------------------------------------------
// Gold path score: one wave per batch, lane-strided over T.
// ---------------------------------------------------------------------------
__global__ __launch_bounds__(256)
void crf_gold_kernel(const float* __restrict__ em,
                     const float* __restrict__ trans,
                     const float* __restrict__ mask,
                     const int* __restrict__ tags,
                     float* __restrict__ gold_out) {
  const int b    = (blockIdx.x * blockDim.x + threadIdx.x) >> 5;
  const int lane = threadIdx.x & 31;
  if (b >= B_) return;

  float partial = 0.f, cnt = 0.f;
  for (int t = lane; t < T_; t += 32) {
    float m = mask[b * T_ + t];
    cnt += m;
    if (t >= 1) {
      int tg = tags[b * T_ + t];
      int tp = tags[b * T_ + t - 1];
      partial += m * (em[(b * T_ + t) * L_ + tg] + trans[tp * L_ + tg]);
    }
  }
#pragma unroll
  for (int off = 16; off >= 1; off >>= 1) {
    partial += __shfl_xor(partial, off, 32);
    cnt     += __shfl_xor(cnt, off, 32);
  }
  if (lane == 0) {
    int t0 = tags[b * T_];
    float sc = trans[BOS_ * L_ + t0] + em[(b * T_) * L_ + t0] + partial;
    int last = (int)(cnt + 0.5f) - 1;        // mask is a contiguous 0/1 prefix
    int tl = tags[b * T_ + last];
    sc += trans[tl * L_ + EOS_];
    gold_out[b] = sc;
  }
}

// ---------------------------------------------------------------------------
// Deterministic final reduction: sum(fwd - gold) -> scalar.
// ---------------------------------------------------------------------------
__global__ __launch_bounds__(256)
void crf_reduce_kernel(const float* __restrict__ fwd,
                       const float* __restrict__ gold,
                       float* __restrict__ out) {
  __shared__ float red[256];
  int tid = threadIdx.x;
  float s = (fwd[tid] - gold[tid]) + (fwd[tid + 256] - gold[tid + 256]);
  red[tid] = s;
  __syncthreads();
  for (int off = 128; off > 0; off >>= 1) {
    if (tid < off) red[tid] += red[tid + off];
    __syncthreads();
  }
  if (tid == 0) out[0] = red[0];
}

extern "C" void kernel_launch(void* const* d_in, const int* in_sizes, int n_in,
                              void* d_out, int out_size, void* d_ws, size_t ws_size,
                              hipStream_t stream) {
  (void)in_sizes; (void)n_in; (void)out_size; (void)ws_size;
  const float* em    = (const float*)d_in[0];   // (512,512,64) f32
  const float* trans = (const float*)d_in[1];   // (64,64) f32
  const float* mask  = (const float*)d_in[2];   // (512,512) f32
  const int*   tags  = (const int*)d_in[3];     // (512,512) i32

  float* fwd  = (float*)d_ws;                   // 512 floats
  float* gold = fwd + B_;                       // 512 floats

  crf_forward_kernel<<<dim3(B_ / 16), dim3(32), 0, stream>>>(em, trans, mask, fwd);
  crf_gold_kernel<<<dim3(B_ / 8), dim3(256), 0, stream>>>(em, trans, mask, tags, gold);
  crf_reduce_kernel<<<dim3(1), dim3(256), 0, stream>>>(fwd, gold, (float*)d_out);
}